// SynthesizerTrn_5403068858972
// MI455X (gfx1250) — compile-verified
//
#include <hip/hip_runtime.h>

#define B_   16
#define C_   192
#define TX_  512
#define TY_  2048
#define SEG_ 32
#define NEG_INF_ (-1e9f)

typedef __attribute__((ext_vector_type(2))) float v2f;
typedef __attribute__((ext_vector_type(8))) float v8f;

#if defined(__has_builtin)
#  if __has_builtin(__builtin_amdgcn_global_load_async_to_lds_b32)
#    define HAVE_ASYNC_LDS 1
#  endif
#endif

typedef __attribute__((address_space(1))) int* gptr_i32;
typedef __attribute__((address_space(3))) int* lptr_i32;

// ---------------- prep: S = exp(-2*logs), SM = S * m ----------------
__global__ __launch_bounds__(256) void prep_k(
    const float* __restrict__ m_p, const float* __restrict__ logs_p,
    float* __restrict__ S, float* __restrict__ SM)
{
  size_t i = (size_t)blockIdx.x * 256 + threadIdx.x;
  const size_t n = (size_t)B_ * C_ * TX_;
  if (i >= n) return;
  float l = logs_p[i];
  float s = __expf(-2.0f * l);
  S[i] = s;
  SM[i] = s * m_p[i];
}

// ---------------- cterm[b,tx] = sum_c (-0.5*s*m^2 - logs) ----------------
__global__ __launch_bounds__(256) void cterm_k(
    const float* __restrict__ m_p, const float* __restrict__ logs_p,
    const float* __restrict__ SM, float* __restrict__ cterm)
{
  int i = blockIdx.x * 256 + threadIdx.x;
  if (i >= B_ * TX_) return;
  int b = i / TX_, tx = i % TX_;
  float acc = 0.f;
  for (int c = 0; c < C_; ++c) {
    size_t o = ((size_t)b * C_ + c) * TX_ + tx;
    acc += -0.5f * SM[o] * m_p[o] - logs_p[o];
  }
  cterm[i] = acc;
}

// ---------------- neg_cent GEMM via V_WMMA_F32_16X16X4_F32 ----------------
// Block = one (b, ty0) group of 32 ty rows; its 8 waves cover tx0 = wave*64.
// The 24KB A-panel z_p[b, :, ty0:ty0+32] is staged once per block into LDS
// (async-to-LDS when available), then both K-regions ( -0.5*z^2 x S and
// z x S*m ) are fused in a single k-loop reusing the same LDS reads:
// 16 wmma per (4 ds_load + 16 global_load) per k-step.
__global__ __launch_bounds__(256) void negcent_gemm_k(
    const float* __restrict__ z_p, const float* __restrict__ S,
    const float* __restrict__ SM, const float* __restrict__ cterm,
    float* __restrict__ neg)
{
  __shared__ float ldsA[C_ * 32];              // 24 KB: [c][ty_local]

  const int lane = threadIdx.x & 31;
  const int wave = threadIdx.x >> 5;
  const int half = lane >> 4;                  // selects K pair {0,1} vs {2,3}
  const int lr   = lane & 15;

  const int TYS = TY_ / 32;                    // 64 ty groups per batch
  const int b   = blockIdx.x / TYS;
  const int ty0 = (blockIdx.x % TYS) * 32;
  const int tx0 = wave * 64;

  // ---- cooperative stage of the A panel ----
  {
    const float* gsrc = z_p + (size_t)b * C_ * TY_ + ty0;   // row c at + c*TY_
    for (int f = threadIdx.x; f < C_ * 32; f += 256) {
      int c = f >> 5, tyl = f & 31;
#if defined(HAVE_ASYNC_LDS)
      __builtin_amdgcn_global_load_async_to_lds_b32(
          (gptr_i32)(gsrc + (size_t)c * TY_ + tyl),
          (lptr_i32)&ldsA[f], 0, 0);
#else
      ldsA[f] = gsrc[(size_t)c * TY_ + tyl];
#endif
    }
#if defined(HAVE_ASYNC_LDS)
#  if __has_builtin(__builtin_amdgcn_s_wait_asynccnt)
    __builtin_amdgcn_s_wait_asynccnt(0);
#  else
    asm volatile("s_wait_asynccnt 0" ::: "memory");
#  endif
#endif
    __syncthreads();
  }

  // B source: S/SM[b, c, tx]
  const size_t bb0 = ((size_t)b * C_ + 2 * half) * TX_ + tx0 + lr;

  v8f acc[2][4] = {};

  for (int i = 0; i < C_ / 4; ++i) {
    const int c0 = 4 * i + 2 * half;
    // A fragments from LDS (shared z panel)
    float z0a = ldsA[c0 * 32 + lr];
    float z0b = ldsA[(c0 + 1) * 32 + lr];
    float z1a = ldsA[c0 * 32 + 16 + lr];
    float z1b = ldsA[(c0 + 1) * 32 + 16 + lr];
    v2f a0q; a0q.x = -0.5f * z0a * z0a; a0q.y = -0.5f * z0b * z0b;
    v2f a1q; a1q.x = -0.5f * z1a * z1a; a1q.y = -0.5f * z1b * z1b;
    v2f a0l; a0l.x = z0a; a0l.y = z0b;
    v2f a1l; a1l.x = z1a; a1l.y = z1b;

    size_t ob = (size_t)(4 * i) * TX_;
#pragma unroll
    for (int t = 0; t < 4; ++t) {
      const float* ps = S + bb0 + 16 * t;
      const float* pm = SM + bb0 + 16 * t;
      v2f bs; bs.x = ps[ob]; bs.y = ps[ob + TX_];
      v2f bm; bm.x = pm[ob]; bm.y = pm[ob + TX_];
      acc[0][t] = __builtin_amdgcn_wmma_f32_16x16x4_f32(
          false, a0q, false, bs, (short)0, acc[0][t], false, false);
      acc[0][t] = __builtin_amdgcn_wmma_f32_16x16x4_f32(
          false, a0l, false, bm, (short)0, acc[0][t], false, false);
      acc[1][t] = __builtin_amdgcn_wmma_f32_16x16x4_f32(
          false, a1q, false, bs, (short)0, acc[1][t], false, false);
      acc[1][t] = __builtin_amdgcn_wmma_f32_16x16x4_f32(
          false, a1l, false, bm, (short)0, acc[1][t], false, false);
    }
  }

  // ---- store: C/D layout: VGPR r -> M = r + 8*half, N = lr ----
#pragma unroll
  for (int t = 0; t < 4; ++t) {
    int col = tx0 + 16 * t + lr;
    float ct = cterm[b * TX_ + col];
#pragma unroll
    for (int g = 0; g < 2; ++g) {
#pragma unroll
      for (int r = 0; r < 8; ++r) {
        int row = ty0 + 16 * g + r + 8 * half;
        neg[((size_t)b * TY_ + row) * TX_ + col] = acc[g][t][r] + ct;
      }
    }
  }
}

// ---------------- fused MAS forward + traceback (one WG per batch) ----------
// Direction bits are ballot-packed into LDS (2048 rows x 16 words = 128 KB of
// the 320KB WGP LDS) so the data-dependent traceback runs at LDS latency.
// Next row of neg_cent is register-prefetched to hide L2 latency behind the
// barrier + ballot work of the current row.
__global__ __launch_bounds__(512) void mas_k(
    const float* __restrict__ neg, const int* __restrict__ xlen,
    const int* __restrict__ ylen, int* __restrict__ idxArr)
{
  extern __shared__ unsigned int smem[];
  float* sv = (float*)smem;                    // 512 floats
  unsigned int* sdirs = smem + 512;            // TY_*16 words

  const int b  = blockIdx.x;
  const int tx = threadIdx.x;
  const int tX = xlen[b];
  const float* nc = neg + (size_t)b * TY_ * TX_;

  float v   = (tx == 0) ? 0.f : NEG_INF_;
  float cur = nc[tx];                          // row 0
  for (int ty = 0; ty < TY_; ++ty) {
    float nxt = (ty + 1 < TY_) ? nc[(size_t)(ty + 1) * TX_ + tx] : 0.f;  // prefetch
    sv[tx] = v;
    __syncthreads();
    float vprev = (tx > 0) ? sv[tx - 1] : NEG_INF_;
    bool dirn = vprev > v;                     // best predecessor is tx-1
    unsigned int mask = __builtin_amdgcn_ballot_w32(dirn);
    if ((tx & 31) == 0) sdirs[ty * 16 + (tx >> 5)] = mask;
    float vnew = cur + fmaxf(v, vprev);
    vnew = (tx < tX) ? vnew : NEG_INF_;
    vnew = fmaxf(vnew, NEG_INF_);
    v = vnew;
    cur = nxt;
    __syncthreads();
  }

  if (tx == 0) {
    const int tY = ylen[b];
    int idx = 0;
    for (int j = TY_ - 1; j >= 0; --j) {
      bool active = j < tY;
      if (j == tY - 1) idx = tX - 1;
      idxArr[b * TY_ + j] = active ? idx : -1;
      if (active) {
        unsigned int w = sdirs[j * 16 + (idx >> 5)];
        idx -= (int)((w >> (idx & 31)) & 1u);
      }
    }
  }
}

// ---------------- attn one-hot expansion ----------------
__global__ __launch_bounds__(256) void attn_k(
    const int* __restrict__ idxArr, float* __restrict__ attn)
{
  size_t i = (size_t)blockIdx.x * 256 + threadIdx.x;
  const size_t n = (size_t)B_ * TY_ * TX_;
  if (i >= n) return;
  int tx = (int)(i % TX_);
  size_t bty = i / TX_;                        // b*TY + ty
  attn[i] = (idxArr[bty] == tx) ? 1.0f : 0.0f;
}

// ---------------- m_p_a / logs_p_a gathers ----------------
__global__ __launch_bounds__(256) void gather_k(
    const int* __restrict__ idxArr, const float* __restrict__ m_p,
    const float* __restrict__ logs_p, float* __restrict__ o_mpa,
    float* __restrict__ o_lpa)
{
  size_t i = (size_t)blockIdx.x * 256 + threadIdx.x;
  const size_t n = (size_t)B_ * C_ * TY_;
  if (i >= n) return;
  int ty = (int)(i % TY_);
  int c  = (int)((i / TY_) % C_);
  int b  = (int)(i / ((size_t)C_ * TY_));
  int id = idxArr[b * TY_ + ty];
  size_t base = ((size_t)b * C_ + c) * TX_;
  o_mpa[i] = (id >= 0) ? m_p[base + id] : 0.0f;
  o_lpa[i] = (id >= 0) ? logs_p[base + id] : 0.0f;
}

// ---------------- random slice segments ----------------
__global__ __launch_bounds__(256) void slice_k(
    const float* __restrict__ z, const int* __restrict__ ylen,
    const float* __restrict__ rand_u, float* __restrict__ o_zs,
    float* __restrict__ o_ids)
{
  size_t i = (size_t)blockIdx.x * 256 + threadIdx.x;
  const size_t n = (size_t)B_ * C_ * SEG_;
  if (i >= n) return;
  int t = (int)(i % SEG_);
  int c = (int)((i / SEG_) % C_);
  int b = (int)(i / ((size_t)C_ * SEG_));
  int safe = min(ylen[b], TY_);
  int ids_max = max(safe - SEG_, 0);
  int ids = (int)(rand_u[b] * ((float)ids_max + 1e-8f));
  bool valid = t < (safe - ids);
  o_zs[i] = valid ? z[((size_t)b * C_ + c) * TY_ + ids + t] : 0.0f;
  if (c == 0 && t == 0) o_ids[b] = (float)ids;
}

extern "C" void kernel_launch(void* const* d_in, const int* in_sizes, int n_in,
                              void* d_out, int out_size, void* d_ws, size_t ws_size,
                              hipStream_t stream) {
  const float* z      = (const float*)d_in[0];
  const float* z_p    = (const float*)d_in[1];
  const float* m_p    = (const float*)d_in[2];
  const float* logs_p = (const float*)d_in[3];
  const int*   xlen   = (const int*)d_in[4];
  const int*   ylen   = (const int*)d_in[5];
  const float* randu  = (const float*)d_in[6];

  const size_t BCTX  = (size_t)B_ * C_ * TX_;        // 1,572,864
  const size_t BTYTX = (size_t)B_ * TY_ * TX_;       // 16,777,216
  const size_t BCTY  = (size_t)B_ * C_ * TY_;        // 6,291,456

  // workspace layout
  float* S     = (float*)d_ws;
  float* SM    = S + BCTX;
  float* cterm = SM + BCTX;
  float* neg   = cterm + (size_t)B_ * TX_;
  int*   idxA  = (int*)(neg + BTYTX);

  // output layout (flat float, reference return order)
  float* o_zs   = (float*)d_out;                        // [B,C,SEG]
  float* o_ids  = o_zs + (size_t)B_ * C_ * SEG_;        // [B]
  float* o_attn = o_ids + B_;                           // [B,Ty,Tx]
  float* o_mpa  = o_attn + BTYTX;                       // [B,C,Ty]
  float* o_lpa  = o_mpa + BCTY;                         // [B,C,Ty]

  // 1) elementwise prep of S, SM
  prep_k<<<(unsigned)((BCTX + 255) / 256), 256, 0, stream>>>(m_p, logs_p, S, SM);

  // 2) cterm reduction over C
  cterm_k<<<(B_ * TX_ + 255) / 256, 256, 0, stream>>>(m_p, logs_p, SM, cterm);

  // 3) WMMA f32 GEMM: 1024 blocks, each = one (b, ty0) group, 8 tx-strip waves
  negcent_gemm_k<<<B_ * (TY_ / 32), 256, 0, stream>>>(z_p, S, SM, cterm, neg);

  // 4) fused MAS DP (forward + LDS-resident traceback), one WG per batch
  const unsigned smem_bytes = (512u + (unsigned)TY_ * 16u) * 4u;  // 133,120 B
  mas_k<<<B_, 512, smem_bytes, stream>>>(neg, xlen, ylen, idxA);

  // 5) outputs
  attn_k<<<(unsigned)((BTYTX + 255) / 256), 256, 0, stream>>>(idxA, o_attn);
  gather_k<<<(unsigned)((BCTY + 255) / 256), 256, 0, stream>>>(idxA, m_p, logs_p,
                                                               o_mpa, o_lpa);
  slice_k<<<(unsigned)(((size_t)B_ * C_ * SEG_ + 255) / 256), 256, 0, stream>>>(
      z, ylen, randu, o_zs, o_ids);
}